// SoftDiceLoss_25632364822677
// MI455X (gfx1250) — compile-verified
//
#include <hip/hip_runtime.h>
#include <hip/hip_bf16.h>

typedef float v2f __attribute__((ext_vector_type(2)));
typedef float v8f __attribute__((ext_vector_type(8)));

#define BS 16
#define NC 21
#define HW (512 * 512)
#define THREADS 256
#define BLOCKS_PER_BATCH (HW / 4 / THREADS)   // 256 blocks per batch sample
#define NBLOCKS (BS * BLOCKS_PER_BATCH)       // 4096 blocks total

// sigmoid(x) = 0.5 + 0.5*tanh(x/2): one transcendental per element (v_tanh_f32
// on gfx1250) instead of exp+rcp -> keeps the kernel memory-bound.
__device__ __forceinline__ float fast_sigmoid(float x) {
#if __has_builtin(__builtin_amdgcn_tanhf)
  return __builtin_amdgcn_tanhf(x * 0.5f) * 0.5f + 0.5f;
#else
  float e = __builtin_amdgcn_exp2f(x * -1.442695040888963f);
  return __builtin_amdgcn_rcpf(1.0f + e);
#endif
}

__global__ __launch_bounds__(THREADS) void dice_partial(
    const float* __restrict__ logits, const long long* __restrict__ targets,
    float2* __restrict__ partial) {
  const int b    = blockIdx.x / BLOCKS_PER_BATCH;
  const int blk  = blockIdx.x % BLOCKS_PER_BATCH;
  const int quad = blk * THREADS + threadIdx.x;   // quad-of-pixels within batch
  const int pix  = quad * 4;

  const long long* tp = targets + (size_t)b * HW + pix;
  const int t0 = (int)tp[0], t1 = (int)tp[1], t2 = (int)tp[2], t3 = (int)tp[3];
  const float* lp = logits + (size_t)b * NC * HW + pix;

  float s = 0.f, inter = 0.f;
#pragma unroll
  for (int c = 0; c < NC; ++c) {
    const float4 v = *(const float4*)(lp + (size_t)c * HW);   // global_load_b128
    if (c + 1 < NC) __builtin_prefetch(lp + (size_t)(c + 1) * HW, 0, 0);
    const float p0 = fast_sigmoid(v.x), p1 = fast_sigmoid(v.y);
    const float p2 = fast_sigmoid(v.z), p3 = fast_sigmoid(v.w);
    s += (p0 + p1) + (p2 + p3);
    inter += (t0 == c ? p0 : 0.f) + (t1 == c ? p1 : 0.f) +
             (t2 == c ? p2 : 0.f) + (t3 == c ? p3 : 0.f);
  }

  // ---- wave32 reduction via V_WMMA_F32_16X16X4_F32 ----
  // A row 0 gets weight 0.5 in all 4 K slots (lanes 0 and 16 hold row 0's K
  // pairs). B carries the per-lane partial duplicated in both B VGPRs, so
  // column n's 4 K-values are {p_n, p_n, p_{n+16}, p_{n+16}} in some order.
  // => D[0,n] = 0.5 * 2 * (p_n + p_{n+16}) = p_n + p_{n+16}, robust to the
  // exact K-slot ordering. Row 0 of D lands in D-VGPR0, lanes 0..15.
  const int lane = threadIdx.x & 31;
  const float sel = (lane == 0 || lane == 16) ? 0.5f : 0.f;
  v2f A;  A[0] = sel;   A[1] = sel;
  v2f Bs; Bs[0] = s;    Bs[1] = s;
  v2f Bi; Bi[0] = inter; Bi[1] = inter;
  v8f cz = {};
  v8f Ds = __builtin_amdgcn_wmma_f32_16x16x4_f32(
      false, A, false, Bs, (short)0, cz, false, false);
  v8f Di = __builtin_amdgcn_wmma_f32_16x16x4_f32(
      false, A, false, Bi, (short)0, cz, false, false);
  float rs = Ds[0];  // lanes 0..15: s_n + s_{n+16}; lanes 16..31: 0
  float ri = Di[0];
#pragma unroll
  for (int off = 8; off >= 1; off >>= 1) {
    rs += __shfl_xor(rs, off, 32);
    ri += __shfl_xor(ri, off, 32);
  }

  __shared__ float2 wpart[THREADS / 32];
  if (lane == 0) wpart[threadIdx.x >> 5] = make_float2(rs, ri);
  __syncthreads();
  if (threadIdx.x == 0) {
    float a = 0.f, c2 = 0.f;
#pragma unroll
    for (int i = 0; i < THREADS / 32; ++i) { a += wpart[i].x; c2 += wpart[i].y; }
    partial[blockIdx.x] = make_float2(a, c2);   // every block writes its slot
  }
}

// One wave (32 lanes) per batch sample reduces its 256 block partials,
// computes the dice score, then thread 0 combines the 16 scores. Fully
// deterministic (no atomics), graph-replay safe.
__global__ __launch_bounds__(512) void dice_finalize(
    const float2* __restrict__ partial, float* __restrict__ out) {
  const int wave = threadIdx.x >> 5;  // 0..15 == batch index
  const int lane = threadIdx.x & 31;
  float s = 0.f, it = 0.f;
#pragma unroll
  for (int j = lane; j < BLOCKS_PER_BATCH; j += 32) {
    const float2 p = partial[wave * BLOCKS_PER_BATCH + j];
    s += p.x; it += p.y;
  }
#pragma unroll
  for (int off = 16; off >= 1; off >>= 1) {
    s  += __shfl_xor(s, off, 32);
    it += __shfl_xor(it, off, 32);
  }
  __shared__ float score[BS];
  if (lane == 0)
    score[wave] = 2.0f * (it + 1.0f) / (s + (float)HW + 1.0f);
  __syncthreads();
  if (threadIdx.x == 0) {
    float t = 0.f;
#pragma unroll
    for (int i = 0; i < BS; ++i) t += score[i];
    out[0] = 1.0f - t / (float)BS;
  }
}

extern "C" void kernel_launch(void* const* d_in, const int* in_sizes, int n_in,
                              void* d_out, int out_size, void* d_ws, size_t ws_size,
                              hipStream_t stream) {
  const float* logits      = (const float*)d_in[0];      // f32 [16,21,512,512]
  const long long* targets = (const long long*)d_in[1];  // int64 [16,512,512]
  float2* partial = (float2*)d_ws;                       // 4096 * 8 B = 32 KB

  dice_partial<<<NBLOCKS, THREADS, 0, stream>>>(logits, targets, partial);
  dice_finalize<<<1, 512, 0, stream>>>(partial, (float*)d_out);

  (void)in_sizes; (void)n_in; (void)out_size; (void)ws_size;
}